// PytorchFFTConv_3126736191554
// MI455X (gfx1250) — compile-verified
//
#include <hip/hip_runtime.h>
#include <hip/hip_bf16.h>

// FFT-based circular convolution on MI455X (gfx1250).
// One 256-thread workgroup per (b,h) row. 4096-point complex FFT done as
// three radix-16 stages; each stage = W16[16x16] x X[16x256] complex matmul
// implemented with V_WMMA_F32_16X16X4_F32 (K=16 via 4 chained WMMAs).
// Row staging (global -> LDS) uses the Tensor Data Mover (tensor_load_to_lds,
// TENSORcnt) so the DMA overlaps twiddle-table build / LDS copy passes.
// Bandwidth-bound: ~214 MB HBM traffic total, compute ~30 GFLOP.

typedef float v2f __attribute__((ext_vector_type(2)));
typedef float v8f __attribute__((ext_vector_type(8)));
typedef unsigned int u32x4 __attribute__((ext_vector_type(4)));
typedef int i32x4 __attribute__((ext_vector_type(4)));
typedef int i32x8 __attribute__((ext_vector_type(8)));

#define L4096 4096
#define TWO_PI 6.28318530717958647692f

#if defined(__gfx1250__) && __has_builtin(__builtin_amdgcn_tensor_load_to_lds) && \
    __has_builtin(__builtin_amdgcn_s_wait_tensorcnt)
#define USE_TDM 1
#else
#define USE_TDM 0
#endif

// base-16 digit reversal of a 3-digit index (swap outer digits)
__device__ __forceinline__ int rev16(int p) {
    return ((p & 15) << 8) | (p & 0xF0) | (p >> 8);
}

#if USE_TDM
// Low 32 bits of a generic address to an LDS object = LDS byte offset
// (aperture lives entirely in the high 32 bits; ISA 10.2).
__device__ __forceinline__ unsigned lds_byte_off(const void* p) {
    return (unsigned)(uintptr_t)p;
}

// TDM: DMA one contiguous row of `nelem` f32 from global to LDS.
// D# per CDNA5 ISA 8.3/8.4: group0={count=1, lds_addr, global_addr, type=2},
// group1={data_size=4B, tensor_dim0=nelem, tensor_dim1=1, tile_dim0=nelem,
// tensor_dim0_stride=nelem}; groups 2/3 zero (1-D tile). Tracked by TENSORcnt.
// This toolchain exposes the 6-arg builtin (clang-23 / therock form):
// (g0 u32x4, g1 i32x8, g2 i32x4, g3 i32x4, i32x8, cpol).
__device__ __forceinline__ void tdm_load_row_f32(unsigned lds_off,
                                                 const float* gptr, int nelem) {
    const unsigned long long ga = (unsigned long long)(uintptr_t)gptr;
    u32x4 g0;
    g0[0] = 1u;                                   // count=1 (valid user D#)
    g0[1] = lds_off;                              // lds_addr (bytes)
    g0[2] = (unsigned)(ga & 0xFFFFFFFFu);         // global_addr[31:0]
    g0[3] = (unsigned)((ga >> 32) & 0x01FFFFFFu)  // global_addr[56:32]
            | (2u << 30);                         // type=2 ("image")
    i32x8 g1;
    g1[0] = (int)(2u << 16);                      // data_size=2 -> 4 bytes
    g1[1] = (int)(((unsigned)nelem & 0xFFFFu) << 16);        // tensor_dim0[15:0]
    g1[2] = (int)((((unsigned)nelem) >> 16) | (1u << 16));   // dim0 hi | tensor_dim1=1
    g1[3] = (int)(((unsigned)nelem) << 16);       // tile_dim0 (16b, 4096 fits)
    g1[4] = 0;                                    // tile_dim1/2 unused (1-D)
    g1[5] = nelem;                                // tensor_dim0_stride[31:0]
    g1[6] = 0;                                    // stride hi | dim1_stride lo
    g1[7] = 0;
    const i32x4 z4 = {0, 0, 0, 0};
    const i32x8 z8 = {0, 0, 0, 0, 0, 0, 0, 0};
    __builtin_amdgcn_tensor_load_to_lds(g0, g1, z4, z4, z8, 0);
}
#endif

// One radix-16 stage: OUT[16x256] = W16 x IN[16x256] (complex), then twiddle.
// Stage S=1 contracts digit n2 (stride 256), S=2 digit n1 (stride 16),
// S=3 digit n0 (stride 1). sgn=+1 forward, -1 inverse (conjugate everything).
// tw[m] = exp(-2*pi*i*m/4096) stored as (twr, twi).
template <int S>
__device__ __forceinline__ void fft_stage(const float* iR, const float* iI,
                                          float* oR, float* oI,
                                          const float* twr, const float* twi,
                                          float sgn, int tid) {
    const int lane  = tid & 31;
    const int wave  = tid >> 5;        // 8 waves
    const int row   = lane & 15;       // M for A, N for B/C
    const int half  = lane >> 4;
    const int khalf = half << 1;

    // Build A operands: W16 real part, imag part, negated imag part.
    // A 16x4 f32 layout: lanes 0-15 hold K=4q,4q+1; lanes 16-31 hold K=4q+2,4q+3.
    v2f Wr[4], Wi[4], Wm[4];
#pragma unroll
    for (int q = 0; q < 4; ++q) {
        const int k0 = 4 * q + khalf;
        const int i0 = ((row * k0) & 15) << 8;        // 256*row*k0 mod 4096
        const int i1 = ((row * (k0 + 1)) & 15) << 8;
        Wr[q].x = twr[i0];        Wr[q].y = twr[i1];
        Wi[q].x = sgn * twi[i0];  Wi[q].y = sgn * twi[i1];
        Wm[q].x = -Wi[q].x;       Wm[q].y = -Wi[q].y;
    }

#pragma unroll
    for (int tt = 0; tt < 2; ++tt) {
        const int t = wave * 2 + tt;   // column tile 0..15, uniform per wave

        // B operands: X tile, 4x16 chunks, same lane-half K split as A.
        v2f Xr[4], Xi[4];
#pragma unroll
        for (int q = 0; q < 4; ++q) {
            const int kk0 = 4 * q + khalf, kk1 = kk0 + 1;
            int a0, a1;
            if (S == 1)      { a0 = 256 * kk0 + 16 * t + row; a1 = 256 * kk1 + 16 * t + row; }
            else if (S == 2) { a0 = 256 * t + 16 * kk0 + row; a1 = 256 * t + 16 * kk1 + row; }
            else             { a0 = 256 * t + 16 * row + kk0; a1 = 256 * t + 16 * row + kk1; }
            Xr[q].x = iR[a0]; Xr[q].y = iR[a1];
            Xi[q].x = iI[a0]; Xi[q].y = iI[a1];
        }

        // Yr = Wr*Xr + (-Wi)*Xi ; Yi = Wr*Xi + Wi*Xr  (K=16 via 4 chained WMMAs)
        v8f accR = {}; v8f accI = {};
#pragma unroll
        for (int q = 0; q < 4; ++q)
            accR = __builtin_amdgcn_wmma_f32_16x16x4_f32(false, Wr[q], false, Xr[q],
                                                         (short)0, accR, false, false);
#pragma unroll
        for (int q = 0; q < 4; ++q)
            accR = __builtin_amdgcn_wmma_f32_16x16x4_f32(false, Wm[q], false, Xi[q],
                                                         (short)0, accR, false, false);
#pragma unroll
        for (int q = 0; q < 4; ++q)
            accI = __builtin_amdgcn_wmma_f32_16x16x4_f32(false, Wr[q], false, Xi[q],
                                                         (short)0, accI, false, false);
#pragma unroll
        for (int q = 0; q < 4; ++q)
            accI = __builtin_amdgcn_wmma_f32_16x16x4_f32(false, Wi[q], false, Xr[q],
                                                         (short)0, accI, false, false);

        // C/D layout: VGPR r holds (M = r + 8*half, N = lane&15).
        const int c = row;
#pragma unroll
        for (int r = 0; r < 8; ++r) {
            const int a = r + 8 * half;  // output digit (k2/k1/k0 for S=1/2/3)
            int m;
            if (S == 1)      m = 16 * t * a;           // w256^(n1*k2), n1=t
            else if (S == 2) m = c * (t + 16 * a);     // w4096^(n0*(k2+16*k1))
            else             m = 0;
            const float fr = twr[m], fi = sgn * twi[m];
            const float cr = accR[r], ci = accI[r];
            const float yr = cr * fr - ci * fi;
            const float yi = cr * fi + ci * fr;
            int oaddr;
            if (S == 1)      oaddr = 256 * a + 16 * t + c;
            else if (S == 2) oaddr = 256 * t + 16 * a + c;
            else             oaddr = 256 * t + 16 * c + a;
            oR[oaddr] = yr; oI[oaddr] = yi;
        }
    }
}

__global__ __launch_bounds__(256) void fftconv4096_kernel(
    const float* __restrict__ x, const float* __restrict__ k,
    float* __restrict__ y, int H) {
    __shared__ float twr[L4096], twi[L4096];   // exp(-2*pi*i*m/4096)
    __shared__ float Asr[L4096], Asi[L4096];   // ping
    __shared__ float Bsr[L4096], Bsi[L4096];   // pong
    __shared__ float Ksr[L4096], Ksi[L4096];   // FFT(k row), natural order

    const int tid = threadIdx.x;
    const int bh  = blockIdx.x;
    const int h   = bh % H;
    const float* xrow = x + (size_t)bh * L4096;
    const float* krow = k + (size_t)h  * L4096;
    float*       yrow = y + (size_t)bh * L4096;

#if USE_TDM
    // Kick off k-row DMA first so it overlaps the twiddle-table build.
    if (tid < 32) tdm_load_row_f32(lds_byte_off(&Asr[0]), krow, L4096);
#endif

    // Twiddle table (accurate libm sincos, once per block).
    for (int j = tid; j < L4096; j += 256) {
        float s, c;
        sincosf((float)j * (-TWO_PI / (float)L4096), &s, &c);
        twr[j] = c; twi[j] = s;
    }
#if USE_TDM
    for (int j = tid; j < L4096; j += 256) Asi[j] = 0.0f;
    __builtin_amdgcn_s_wait_tensorcnt(0);      // no-op for waves with cnt==0
#else
    for (int j = tid; j < L4096; j += 256) { Asr[j] = krow[j]; Asi[j] = 0.0f; }
#endif
    __syncthreads();

    // FFT(k): three radix-16 WMMA stages, result digit-reversed in B.
    fft_stage<1>(Asr, Asi, Bsr, Bsi, twr, twi, 1.0f, tid); __syncthreads();
    fft_stage<2>(Bsr, Bsi, Asr, Asi, twr, twi, 1.0f, tid); __syncthreads();
    fft_stage<3>(Asr, Asi, Bsr, Bsi, twr, twi, 1.0f, tid); __syncthreads();

#if USE_TDM
    // A is dead now: start the x-row DMA, overlap it with the K un-reverse copy.
    if (tid < 32) tdm_load_row_f32(lds_byte_off(&Asr[0]), xrow, L4096);
#endif
    for (int j = tid; j < L4096; j += 256) {
        const int r = rev16(j);
        Ksr[j] = Bsr[r]; Ksi[j] = Bsi[r];      // un-digit-reverse -> natural
    }
#if USE_TDM
    for (int j = tid; j < L4096; j += 256) Asi[j] = 0.0f;
    __builtin_amdgcn_s_wait_tensorcnt(0);
#else
    {
        const float4* x4 = reinterpret_cast<const float4*>(xrow);
        for (int j = tid; j < L4096 / 4; j += 256) {
            const float4 v = x4[j];
            const int b = 4 * j;
            Asr[b] = v.x; Asr[b + 1] = v.y; Asr[b + 2] = v.z; Asr[b + 3] = v.w;
            Asi[b] = 0.0f; Asi[b + 1] = 0.0f; Asi[b + 2] = 0.0f; Asi[b + 3] = 0.0f;
        }
    }
#endif
    __syncthreads();

    // FFT(x)
    fft_stage<1>(Asr, Asi, Bsr, Bsi, twr, twi, 1.0f, tid); __syncthreads();
    fft_stage<2>(Bsr, Bsi, Asr, Asi, twr, twi, 1.0f, tid); __syncthreads();
    fft_stage<3>(Asr, Asi, Bsr, Bsi, twr, twi, 1.0f, tid); __syncthreads();

    // Pointwise complex product (un-reversing X on the fly), into A (natural).
    for (int j = tid; j < L4096; j += 256) {
        const int r = rev16(j);
        const float xr = Bsr[r], xi = Bsi[r];
        const float kr = Ksr[j], ki = Ksi[j];
        Asr[j] = xr * kr - xi * ki;
        Asi[j] = xr * ki + xi * kr;
    }
    __syncthreads();

    // Inverse FFT (conjugate twiddles), result digit-reversed in B.
    fft_stage<1>(Asr, Asi, Bsr, Bsi, twr, twi, -1.0f, tid); __syncthreads();
    fft_stage<2>(Bsr, Bsi, Asr, Asi, twr, twi, -1.0f, tid); __syncthreads();
    fft_stage<3>(Asr, Asi, Bsr, Bsi, twr, twi, -1.0f, tid); __syncthreads();

    // y[l] = Re(B[rev(l)]) / N  (coalesced global stores)
    for (int j = tid; j < L4096; j += 256)
        yrow[j] = Bsr[rev16(j)] * (1.0f / (float)L4096);
}

extern "C" void kernel_launch(void* const* d_in, const int* in_sizes, int n_in,
                              void* d_out, int out_size, void* d_ws, size_t ws_size,
                              hipStream_t stream) {
    (void)n_in; (void)d_ws; (void)ws_size; (void)out_size;
    const float* x = (const float*)d_in[0];   // [B,H,4096] f32
    const float* k = (const float*)d_in[1];   // [H,4096]   f32
    float*       y = (float*)d_out;           // [B,H,4096] f32
    const int rows = in_sizes[0] / L4096;     // B*H = 6144
    const int H    = in_sizes[1] / L4096;     // 768
    fftconv4096_kernel<<<rows, 256, 0, stream>>>(x, k, y, H);
}